// RecurrentResonanceModel_55516747268512
// MI455X (gfx1250) — compile-verified
//
#include <hip/hip_runtime.h>
#include <math.h>

typedef float v2f __attribute__((ext_vector_type(2)));
typedef float v4f __attribute__((ext_vector_type(4)));
typedef float v8f __attribute__((ext_vector_type(8)));

#define BASE_RES    0.2f
#define RES_FACTORf 0.72f          /* (1-0.2)*0.9 */
#define NROWS       1024           /* B*E */
#define LATENT      128
#define N_ATOMS     1024
#define SAMPLES     32768
#define HOP         256
#define WIN         512
#define NFRAMES     128

__device__ __forceinline__ float hannw(int w) {
    // periodic hann, length 512
    return 0.5f * (1.0f - cosf(6.283185307179586f * (float)w * (1.0f / 512.0f)));
}

// ---------------------------------------------------------------------------
// Generic fp32 WMMA GEMM:  C[M x N] = A[M x K] * B, B element (k,n) at
// B[k*bsk + n*bsn].  M,N multiples of 64, K multiple of 32.
// Block: 128 threads = 4 waves; block tile 64x64; wave tile 16x64.
//
// LDS layouts engineered so every WMMA operand is one aligned ds_load_b64:
//   As[r][k]        : row stride 36 (mult of 4)  -> (k,k+1) pair 8B aligned
//   Bs[k>>1][2n+k&1]: K-pair interleaved         -> (k,k+1)@col n contiguous
// BVEC: vectorize global B staging (requires bsn == 1).
// ---------------------------------------------------------------------------
template <bool BVEC>
__global__ __launch_bounds__(128)
void gemm_f32_wmma(const float* __restrict__ A, int lda,
                   const float* __restrict__ B, long long bsk, long long bsn,
                   float* __restrict__ C, int ldc, int K)
{
    __shared__ __align__(16) float As[64][36];    // 64 rows x 32 k (+pad)
    __shared__ __align__(16) float Bs[16][130];   // 16 k-pairs x 64 cols x 2 (+pad)

    const int tid   = threadIdx.x;
    const int wave  = tid >> 5;
    const int lane  = tid & 31;
    const int lhalf = lane >> 4;   // 0: lanes 0-15, 1: lanes 16-31
    const int l16   = lane & 15;
    const int row0  = blockIdx.y * 64;
    const int col0  = blockIdx.x * 64;

    v8f acc[4];
#pragma unroll
    for (int t = 0; t < 4; ++t)
#pragma unroll
        for (int i = 0; i < 8; ++i) acc[t][i] = 0.0f;

    for (int k0 = 0; k0 < K; k0 += 32) {
        // ---- stage A tile (64x32), float4 per thread x4 -------------------
#pragma unroll
        for (int i = tid * 4; i < 64 * 32; i += 128 * 4) {
            const int r = i >> 5, k = i & 31;
            const v4f v = *(const v4f*)&A[(size_t)(row0 + r) * lda + (k0 + k)];
            *(v4f*)&As[r][k] = v;            // r*36 + k is 16B aligned
        }
        // ---- stage B tile (32x64) into K-pair-interleaved layout ----------
        if (BVEC) {  // bsn == 1: vectorized global read (the atoms stream)
#pragma unroll
            for (int i = tid * 4; i < 32 * 64; i += 128 * 4) {
                const int k = i >> 6, n = i & 63;
                const v4f v = *(const v4f*)&B[(long long)(k0 + k) * bsk + (col0 + n)];
                float* dst = &Bs[k >> 1][(k & 1)];
                dst[2 * (n + 0)] = v.x;
                dst[2 * (n + 1)] = v.y;
                dst[2 * (n + 2)] = v.z;
                dst[2 * (n + 3)] = v.w;
            }
        } else {
#pragma unroll
            for (int i = tid; i < 32 * 64; i += 128) {
                const int k = i >> 6, n = i & 63;
                Bs[k >> 1][2 * n + (k & 1)] =
                    B[(long long)(k0 + k) * bsk + (long long)(col0 + n) * bsn];
            }
        }
        __syncthreads();

#pragma unroll
        for (int kk = 0; kk < 32; kk += 4) {
            const int ak = kk + (lhalf << 1);              // even
            const v2f a = *(const v2f*)&As[(wave << 4) + l16][ak];
            const int p = ak >> 1;
#pragma unroll
            for (int t = 0; t < 4; ++t) {
                const v2f b = *(const v2f*)&Bs[p][((t << 4) + l16) << 1];
                acc[t] = __builtin_amdgcn_wmma_f32_16x16x4_f32(
                    false, a, false, b, (short)0, acc[t], false, false);
            }
        }
        __syncthreads();
    }

    // store: acc[t][i] -> C[row0 + 16*wave + i + 8*lhalf][col0 + 16*t + l16]
#pragma unroll
    for (int t = 0; t < 4; ++t) {
        const int n = col0 + (t << 4) + l16;
#pragma unroll
        for (int i = 0; i < 8; ++i) {
            const int m = row0 + (wave << 4) + (lhalf << 3) + i;
            C[(size_t)m * ldc + n] = acc[t][i];
        }
    }
}

// ---------------------------------------------------------------------------
// mom: sigmoid affine + cumulative product over F=128; one block per row.
// ---------------------------------------------------------------------------
__global__ __launch_bounds__(128)
void mom_kernel(const float* __restrict__ logits, const float* __restrict__ b_mom,
                float* __restrict__ mom_out)
{
    __shared__ float m[NFRAMES];
    const int row = blockIdx.x;
    const int f   = threadIdx.x;
    float v = logits[row * NFRAMES + f] + b_mom[f];
    float s = 1.0f / (1.0f + expf(-v));
    m[f] = 1e-12f + (BASE_RES + s * RES_FACTORf);
    __syncthreads();
    if (f == 0) {
        float p = 1.0f;
        for (int i = 0; i < NFRAMES; ++i) { p *= m[i]; m[i] = p; }
    }
    __syncthreads();
    mom_out[row * NFRAMES + f] = m[f];
}

// ---------------------------------------------------------------------------
// Row softmax over 1024 atoms, in place. One block (256 thr) per row.
// ---------------------------------------------------------------------------
__global__ __launch_bounds__(256)
void softmax_kernel(float* __restrict__ logits, const float* __restrict__ bias)
{
    __shared__ float red[256];
    const int row = blockIdx.x;
    float* p = logits + (size_t)row * N_ATOMS;

    float v[4];
    float mx = -INFINITY;
#pragma unroll
    for (int j = 0; j < 4; ++j) {
        int c = threadIdx.x + 256 * j;
        v[j] = p[c] + bias[c];
        mx = fmaxf(mx, v[j]);
    }
    red[threadIdx.x] = mx;
    __syncthreads();
    for (int s = 128; s > 0; s >>= 1) {
        if (threadIdx.x < s) red[threadIdx.x] = fmaxf(red[threadIdx.x], red[threadIdx.x + s]);
        __syncthreads();
    }
    mx = red[0];
    __syncthreads();

    float sum = 0.0f;
#pragma unroll
    for (int j = 0; j < 4; ++j) { v[j] = expf(v[j] - mx); sum += v[j]; }
    red[threadIdx.x] = sum;
    __syncthreads();
    for (int s = 128; s > 0; s >>= 1) {
        if (threadIdx.x < s) red[threadIdx.x] += red[threadIdx.x + s];
        __syncthreads();
    }
    const float inv = 1.0f / red[0];
#pragma unroll
    for (int j = 0; j < 4; ++j) p[threadIdx.x + 256 * j] = v[j] * inv;
}

// ---------------------------------------------------------------------------
// Per-(row,frame) gain: g = mom / (||hann * res[f*HOP : f*HOP+WIN]|| + 1e-8).
// One block (128 thr) per (row*128 + f).
// ---------------------------------------------------------------------------
__global__ __launch_bounds__(128)
void frame_gain_kernel(const float* __restrict__ res, const float* __restrict__ mom,
                       float* __restrict__ g)
{
    __shared__ float red[128];
    const int idx = blockIdx.x;          // row*NFRAMES + f
    const int row = idx >> 7;
    const int f   = idx & (NFRAMES - 1);
    const float* base = res + (size_t)row * SAMPLES;

    float ss = 0.0f;
#pragma unroll
    for (int j = 0; j < 4; ++j) {
        int w = threadIdx.x + 128 * j;
        int s = f * HOP + w;
        float x = (s < SAMPLES) ? base[s] : 0.0f;   // zero pad at tail
        x *= hannw(w);
        ss += x * x;
    }
    red[threadIdx.x] = ss;
    __syncthreads();
    for (int s = 64; s > 0; s >>= 1) {
        if (threadIdx.x < s) red[threadIdx.x] += red[threadIdx.x + s];
        __syncthreads();
    }
    if (threadIdx.x == 0)
        g[idx] = mom[idx] / (sqrtf(red[0]) + 1e-8f);
}

// ---------------------------------------------------------------------------
// Fused window + overlap-add: out[s] = res[s] * (hann(w)*g[f] + hann(w+256)*g[f-1])
// ---------------------------------------------------------------------------
__global__ __launch_bounds__(256)
void ola_kernel(const float* __restrict__ res, const float* __restrict__ g,
                float* __restrict__ out)
{
    const size_t i = (size_t)blockIdx.x * 256 + threadIdx.x;
    const int row = (int)(i >> 15);
    const int s   = (int)(i & (SAMPLES - 1));
    const int f   = s >> 8;
    const int w   = s & (HOP - 1);
    float gain = hannw(w) * g[row * NFRAMES + f];
    if (f > 0) gain += hannw(w + HOP) * g[row * NFRAMES + f - 1];
    out[i] = res[i] * gain;
}

// ---------------------------------------------------------------------------
extern "C" void kernel_launch(void* const* d_in, const int* in_sizes, int n_in,
                              void* d_out, int out_size, void* d_ws, size_t ws_size,
                              hipStream_t stream)
{
    (void)in_sizes; (void)n_in; (void)out_size; (void)ws_size;
    const float* x     = (const float*)d_in[0];   // (16,64,128)
    const float* W_sel = (const float*)d_in[1];   // (1024,128)
    const float* b_sel = (const float*)d_in[2];   // (1024,)
    const float* W_mom = (const float*)d_in[3];   // (128,128)
    const float* b_mom = (const float*)d_in[4];   // (128,)
    const float* atoms = (const float*)d_in[5];   // (1024,32768)

    float* out     = (float*)d_out;                       // (B,E,32768)
    float* mom_out = out + (size_t)NROWS * SAMPLES;       // (B,E,128) second output

    char*  ws      = (char*)d_ws;
    float* res     = (float*)(ws);                                   // 134217728 B
    float* logits  = (float*)(ws + (size_t)134217728);               //   4194304 B
    float* mlogits = (float*)(ws + (size_t)138412032);               //    524288 B
    float* g       = (float*)(ws + (size_t)138936320);               //    524288 B

    const dim3 blk(128);

    // 1) momentum logits: (1024x128) = x(1024x128) * W_mom^T  (B: bsk=1, bsn=128)
    gemm_f32_wmma<false><<<dim3(NFRAMES / 64, NROWS / 64), blk, 0, stream>>>(
        x, LATENT, W_mom, 1LL, (long long)LATENT, mlogits, NFRAMES, LATENT);

    // 2) sigmoid + cumprod -> mom output (tail of d_out)
    mom_kernel<<<NROWS, NFRAMES, 0, stream>>>(mlogits, b_mom, mom_out);

    // 3) selection logits: (1024x1024) = x * W_sel^T
    gemm_f32_wmma<false><<<dim3(N_ATOMS / 64, NROWS / 64), blk, 0, stream>>>(
        x, LATENT, W_sel, 1LL, (long long)LATENT, logits, N_ATOMS, LATENT);

    // 4) row softmax (in place -> sel)
    softmax_kernel<<<NROWS, 256, 0, stream>>>(logits, b_sel);

    // 5) big GEMM: res(1024x32768) = sel(1024x1024) * atoms(1024x32768)
    gemm_f32_wmma<true><<<dim3(SAMPLES / 64, NROWS / 64), blk, 0, stream>>>(
        logits, N_ATOMS, atoms, (long long)SAMPLES, 1LL, res, SAMPLES, N_ATOMS);

    // 6) per-frame gains
    frame_gain_kernel<<<NROWS * NFRAMES, 128, 0, stream>>>(res, mom_out, g);

    // 7) fused window/normalize/momentum/overlap-add
    ola_kernel<<<(NROWS * SAMPLES) / 256, 256, 0, stream>>>(res, g, out);
}